// SparseStochasticLatent_22419729286054
// MI455X (gfx1250) — compile-verified
//
#include <hip/hip_runtime.h>
#include <math.h>

#define D_DIM 512
#define M_DIM 256
#define T_DIM 4096
#define LDA   257          // padded row stride (mod 64 banks == 1)
#define NB    16
#define NT    16           // 16x16 tiles per dim
#define NTHREADS 512
#define NWAVES   16
#define JITTER 1e-6f

typedef float v2f __attribute__((ext_vector_type(2)));
typedef float v8f __attribute__((ext_vector_type(8)));

// LDS layout (floats): Kt[256*257] | erow[256] | yv[256] | dtmp[16*17] | wtmp[16*16*17] | red[32]
#define SM_KT    0
#define SM_EROW  (M_DIM * LDA)
#define SM_YV    (SM_EROW + M_DIM)
#define SM_DTMP  (SM_YV + M_DIM)
#define SM_WTMP  (SM_DTMP + 16 * 17)
#define SM_RED   (SM_WTMP + NWAVES * 16 * 17)
#define SM_TOTAL (SM_RED + 32)

__device__ __forceinline__ void wave_fence() { __builtin_amdgcn_wave_barrier(); }

__device__ __forceinline__ float block_reduce(float v, float* red, int lane, int wave) {
  #pragma unroll
  for (int off = 16; off > 0; off >>= 1) v += __shfl_down(v, off, 32);
  if (lane == 0) red[wave] = v;
  __syncthreads();
  if (wave == 0) {
    float r = (lane < NWAVES) ? red[lane] : 0.0f;
    #pragma unroll
    for (int off = 8; off > 0; off >>= 1) r += __shfl_down(r, off, 32);
    if (lane == 0) red[0] = r;
  }
  __syncthreads();
  float result = red[0];
  __syncthreads();
  return result;
}

__global__ void zero_out_kernel(float* out) { out[0] = 0.0f; }

__global__ __launch_bounds__(NTHREADS, 1)
void gp_elbo_kernel(const float* __restrict__ ell_raw,
                    const float* __restrict__ U_mean,
                    const float* __restrict__ postK,
                    float* __restrict__ out) {
  extern __shared__ float smem[];
  float* Kt   = smem + SM_KT;
  float* erow = smem + SM_EROW;
  float* yv   = smem + SM_YV;
  float* dtmp = smem + SM_DTMP;
  float* wtmp = smem + SM_WTMP;
  float* red  = smem + SM_RED;

  const int tid  = threadIdx.x;
  const int lane = tid & 31;
  const int wave = tid >> 5;
  const int d    = blockIdx.x;

  // lane->fragment mappings (CDNA5 f32 WMMA 16x16x4)
  const int am    = lane & 15;          // A row / B col / generic 0..15 index
  const int k0    = (lane >> 4) * 2;    // K sub-offset: lanes 0-15 -> K{0,1}, 16-31 -> K{2,3}
  const int ncol  = lane & 15;          // C/D column
  const int mbase = (lane >> 4) * 8;    // C/D row base: VGPR v -> row mbase+v

  // ---------------- Phase 0: build K (Toeplitz RBF + jitter) in LDS ----------
  float xr  = ell_raw[d];
  float ell = (xr > 20.0f) ? xr : log1pf(__expf(xr));
  const float h = (float)(T_DIM + 2) / (float)(M_DIM - 1);
  const float coef = (h * h) / (2.0f * ell * ell);
  if (tid < M_DIM) {
    float kk = (float)tid;
    erow[tid] = __expf(-kk * kk * coef);
  }
  __syncthreads();
  for (int idx = tid; idx < M_DIM * M_DIM; idx += NTHREADS) {
    int i = idx >> 8, j = idx & 255;
    int dd = i - j; if (dd < 0) dd = -dd;
    Kt[i * LDA + j] = erow[dd] + ((i == j) ? JITTER : 0.0f);
  }
  __syncthreads();

  // ---------------- Phase 1: blocked Cholesky (lower), WMMA trailing updates -
  for (int k = 0; k < NT; ++k) {
    const int kb = k * NB;
    // POTF2 on diagonal block: wave 0
    if (wave == 0) {
      for (int j = 0; j < NB; ++j) {
        float djj = sqrtf(Kt[(kb + j) * LDA + kb + j]);
        float v = 0.0f;
        if (lane < NB && lane >= j) {
          v = Kt[(kb + lane) * LDA + kb + j] / djj;   // lane==j -> djj
          Kt[(kb + lane) * LDA + kb + j] = v;
        }
        wave_fence();
        if (lane < NB) {
          for (int c = j + 1; c < NB; ++c) {
            float lcj = __shfl(v, c, 32);
            if (lane >= c) Kt[(kb + lane) * LDA + kb + c] -= v * lcj;
          }
        }
        wave_fence();
      }
    }
    __syncthreads();
    // TRSM: rows below panel, one row per thread
    {
      int r = kb + NB + tid;
      if (r < M_DIM) {
        float x[NB];
        #pragma unroll
        for (int c = 0; c < NB; ++c) x[c] = Kt[r * LDA + kb + c];
        #pragma unroll
        for (int c = 0; c < NB; ++c) {
          #pragma unroll
          for (int m = 0; m < NB; ++m)
            if (m < c) x[c] -= x[m] * Kt[(kb + c) * LDA + kb + m];
          x[c] /= Kt[(kb + c) * LDA + kb + c];
        }
        #pragma unroll
        for (int c = 0; c < NB; ++c) Kt[r * LDA + kb + c] = x[c];
      }
    }
    __syncthreads();
    // Trailing update: A[BI][BJ] -= P_BI * P_BJ^T  (WMMA f32 16x16x4 chains)
    {
      const int nbr = NT - 1 - k;
      const int ntiles = nbr * (nbr + 1) / 2;
      for (int t = wave; t < ntiles; t += NWAVES) {
        int bj = 0, tt = t;
        while (tt >= nbr - bj) { tt -= nbr - bj; bj++; }
        const int BJ = k + 1 + bj;
        const int BI = BJ + tt;
        v8f c;
        #pragma unroll
        for (int v = 0; v < 8; ++v)
          c[v] = Kt[(BI * NB + mbase + v) * LDA + BJ * NB + ncol];
        #pragma unroll
        for (int q = 0; q < 4; ++q) {
          v2f a, b;
          a.x = -Kt[(BI * NB + am) * LDA + kb + q * 4 + k0];
          a.y = -Kt[(BI * NB + am) * LDA + kb + q * 4 + k0 + 1];
          b.x =  Kt[(BJ * NB + am) * LDA + kb + q * 4 + k0];
          b.y =  Kt[(BJ * NB + am) * LDA + kb + q * 4 + k0 + 1];
          c = __builtin_amdgcn_wmma_f32_16x16x4_f32(false, a, false, b, (short)0, c, false, false);
        }
        #pragma unroll
        for (int v = 0; v < 8; ++v)
          Kt[(BI * NB + mbase + v) * LDA + BJ * NB + ncol] = c[v];
      }
    }
    __syncthreads();
  }

  // ---------------- Phase 2: logdet = 2*sum(log diag(L)) -----------------
  float ldv = 0.0f;
  if (tid < M_DIM) ldv = 2.0f * __logf(Kt[tid * LDA + tid]);
  float det_term = block_reduce(ldv, red, lane, wave);

  // ---------------- Phase 3: mean term = ||L^{-1} u||^2 ------------------
  if (tid < M_DIM) yv[tid] = U_mean[(size_t)d * M_DIM + tid];
  __syncthreads();
  if (wave == 0) {
    for (int j = 0; j < M_DIM; ++j) {
      float yj = yv[j] / Kt[j * LDA + j];
      if (lane == 0) yv[j] = yj;
      wave_fence();
      for (int i = j + 1 + lane; i < M_DIM; i += 32)
        yv[i] -= Kt[i * LDA + j] * yj;
      wave_fence();
    }
  }
  __syncthreads();
  float mtv = 0.0f;
  if (tid < M_DIM) { float yy = yv[tid]; mtv = yy * yy; }
  float mean_term = block_reduce(mtv, red, lane, wave);

  // ---------------- Phase 4: in-place blocked W = L^{-1} ------------------
  for (int ib = 0; ib < NT; ++ib) {
    // invert diagonal block (wave 0) via copy to dtmp
    if (wave == 0) {
      for (int idx = lane; idx < 256; idx += 32) {
        int r = idx >> 4, cc = idx & 15;
        dtmp[r * 17 + cc] = Kt[(ib * NB + r) * LDA + ib * NB + cc];
      }
      wave_fence();
      if (lane < NB) {
        const int cc = lane;
        float wcol[NB];
        #pragma unroll
        for (int r = 0; r < NB; ++r) {
          float val = 0.0f;
          if (r == cc) {
            val = 1.0f / dtmp[r * 17 + r];
          } else if (r > cc) {
            float s = 0.0f;
            #pragma unroll
            for (int m = 0; m < NB; ++m)
              if (m >= cc && m < r) s += dtmp[r * 17 + m] * wcol[m];
            val = -s / dtmp[r * 17 + r];
          }
          wcol[r] = val;
          if (r >= cc) Kt[(ib * NB + r) * LDA + ib * NB + cc] = val;
        }
      }
    }
    __syncthreads();
    // S_j = sum_{m=j}^{ib-1} L[ib][m] * W[m][j]   (wave j handles column j)
    if (wave < ib) {
      const int j = wave;
      v8f s = {0.f,0.f,0.f,0.f,0.f,0.f,0.f,0.f};
      for (int m = j; m < ib; ++m) {
        const bool bd = (m == j);   // W[j][j] is lower-triangular: mask upper
        #pragma unroll
        for (int q = 0; q < 4; ++q) {
          const int kc0 = q * 4 + k0, kc1 = kc0 + 1;
          v2f a, b;
          a.x = Kt[(ib * NB + am) * LDA + m * NB + kc0];
          a.y = Kt[(ib * NB + am) * LDA + m * NB + kc1];
          b.x = (!bd || kc0 >= am) ? Kt[(m * NB + kc0) * LDA + j * NB + am] : 0.0f;
          b.y = (!bd || kc1 >= am) ? Kt[(m * NB + kc1) * LDA + j * NB + am] : 0.0f;
          s = __builtin_amdgcn_wmma_f32_16x16x4_f32(false, a, false, b, (short)0, s, false, false);
        }
      }
      #pragma unroll
      for (int v = 0; v < 8; ++v)
        wtmp[wave * 272 + (mbase + v) * 17 + ncol] = s[v];
    }
    __syncthreads();
    // W[ib][j] = -W[ib][ib] * S_j
    if (wave < ib) {
      const int j = wave;
      v8f w = {0.f,0.f,0.f,0.f,0.f,0.f,0.f,0.f};
      #pragma unroll
      for (int q = 0; q < 4; ++q) {
        const int kc0 = q * 4 + k0, kc1 = kc0 + 1;
        v2f a, b;
        a.x = (kc0 <= am) ? -Kt[(ib * NB + am) * LDA + ib * NB + kc0] : 0.0f;
        a.y = (kc1 <= am) ? -Kt[(ib * NB + am) * LDA + ib * NB + kc1] : 0.0f;
        b.x = wtmp[wave * 272 + kc0 * 17 + am];
        b.y = wtmp[wave * 272 + kc1 * 17 + am];
        w = __builtin_amdgcn_wmma_f32_16x16x4_f32(false, a, false, b, (short)0, w, false, false);
      }
      #pragma unroll
      for (int v = 0; v < 8; ++v)
        Kt[(ib * NB + mbase + v) * LDA + j * NB + ncol] = w[v];
    }
    __syncthreads();
  }

  // ------- Phase 5: Kinv = W^T W tile-by-tile, fused dot with streamed P ---
  float tracc = 0.0f;
  {
    const float* __restrict__ Pd = postK + (size_t)d * (M_DIM * M_DIM);
    for (int t = wave; t < NT * (NT + 1) / 2; t += NWAVES) {
      int I = 0, tt = t;
      while (tt >= NT - I) { tt -= NT - I; I++; }
      const int J = I + tt;            // I <= J
      v8f c = {0.f,0.f,0.f,0.f,0.f,0.f,0.f,0.f};
      for (int m = J; m < NT; ++m) {   // m >= max(I,J)
        const bool ad = (m == I);      // diagonal tile as A: mask upper
        const bool bd = (m == J);      // diagonal tile as B: mask upper
        #pragma unroll
        for (int q = 0; q < 4; ++q) {
          const int kc0 = q * 4 + k0, kc1 = kc0 + 1;
          v2f a, b;
          a.x = (!ad || kc0 >= am) ? Kt[(m * NB + kc0) * LDA + I * NB + am] : 0.0f;
          a.y = (!ad || kc1 >= am) ? Kt[(m * NB + kc1) * LDA + I * NB + am] : 0.0f;
          b.x = (!bd || kc0 >= am) ? Kt[(m * NB + kc0) * LDA + J * NB + am] : 0.0f;
          b.y = (!bd || kc1 >= am) ? Kt[(m * NB + kc1) * LDA + J * NB + am] : 0.0f;
          c = __builtin_amdgcn_wmma_f32_16x16x4_f32(false, a, false, b, (short)0, c, false, false);
        }
      }
      const float wgt = (I == J) ? 1.0f : 2.0f;   // P, Kinv both symmetric
      #pragma unroll
      for (int v = 0; v < 8; ++v) {
        int r  = I * NB + mbase + v;
        int cc = J * NB + ncol;
        float p = __builtin_nontemporal_load(&Pd[r * M_DIM + cc]);
        tracc += wgt * c[v] * p;
      }
    }
  }
  float trace_term = block_reduce(tracc, red, lane, wave);

  // ---------------- Final: -0.5 * nansum over d ----------------------------
  if (tid == 0) {
    float total = trace_term + mean_term + det_term;
    if (!(total != total))             // nansum: skip NaN contributions
      atomicAdd(out, -0.5f * total);
  }
}

extern "C" void kernel_launch(void* const* d_in, const int* in_sizes, int n_in,
                              void* d_out, int out_size, void* d_ws, size_t ws_size,
                              hipStream_t stream) {
  const float* ell_raw = (const float*)d_in[0];
  const float* U_mean  = (const float*)d_in[1];
  const float* postK   = (const float*)d_in[2];
  float* out = (float*)d_out;

  zero_out_kernel<<<1, 1, 0, stream>>>(out);
  const size_t shmem = (size_t)SM_TOTAL * sizeof(float);   // ~277 KB, fits 320 KB CDNA5 LDS
  gp_elbo_kernel<<<D_DIM, NTHREADS, shmem, stream>>>(ell_raw, U_mean, postK, out);
}